// CriterionLP_all_84447646974569
// MI455X (gfx1250) — compile-verified
//
#include <hip/hip_runtime.h>
#include <math.h>
#include <stdint.h>

typedef float v2f __attribute__((ext_vector_type(2)));
typedef float v8f __attribute__((ext_vector_type(8)));

#define B_ROWS   4096
#define N_COLS   16384
#define CK       128
#define MTILE    32                      // rows per block (2 x 16-row WMMA subtiles)
#define NW       8                       // waves per block
#define NBLOCKS  (B_ROWS / MTILE)        // 128
#define NTILES   (N_COLS / 16)           // 1024 column tiles
#define TPW      (NTILES / NW)           // 128 tiles per wave
#define KCH      32                      // K-chunk width (floats)
#define NKC      (CK / KCH)              // 4 chunks per tile
#define NKS      (CK / 4)                // 32 WMMA k-steps per tile
#define BPAD     36                      // padded LDS row stride: n*36 % 64 distinct for n=0..15
#define INV_TEMP 20.0f                   // 1 / 0.05

// one-exp online-softmax merge of (m,s) with a single sample v
__device__ __forceinline__ void online_update(float& m, float& s, float v) {
    const float d = v - m;                    // -inf bootstrap: d=+inf -> e=0 -> s=fma(s,0,1)=1
    const float e = __expf(-fabsf(d));
    s = (d > 0.0f) ? fmaf(s, e, 1.0f) : (s + e);
    m = fmaxf(m, v);
}

// one-exp merge of two (m,s) pairs
__device__ __forceinline__ void pair_merge(float& m, float& s, float mo, float so) {
    const float d = m - mo;
    const float e = __expf(-fabsf(d));
    s = (d >= 0.0f) ? fmaf(so, e, s) : fmaf(s, e, so);
    m = fmaxf(m, mo);
}

union SmemU {
    float a[MTILE * CK];                 // 16 KB  raw A staging (consumed, then reused)
    float b[NW][2][16 * BPAD];           // 36.9 KB per-wave double-buffered B chunks
};

// Fused  sim = feats @ fs^T / temp  ->  online logsumexp + positive sum/count -> per-block loss
__global__ __launch_bounds__(256) void lp_loss_main(
    const float* __restrict__ feats,     // [4096,128]
    const float* __restrict__ fs,        // [16384,128]
    const int*   __restrict__ labels,    // [4096]
    const int*   __restrict__ labels_s,  // [16384]
    float*       __restrict__ block_loss)// [128]
{
    __shared__ SmemU  smem;
    __shared__ v2f    a1f[NKS * 32];           // 8 KB  subtile-1 A fragments, [kstep][lane] order
    __shared__ float4 lds_red[NW][MTILE];      // 4 KB  per-wave per-row (m, s, possum, poscnt)

    const int tid     = threadIdx.x;
    const int wv      = tid >> 5;
    const int lane    = tid & 31;
    const int half    = lane >> 4;             // which 16-lane half
    const int lm      = lane & 15;             // row index (A frag) / column index (B & C frags)
    const int rowbase = blockIdx.x * MTILE;

    // ---- stage raw A tile (coalesced) ----
    for (int idx = tid; idx < MTILE * CK; idx += 256)
        smem.a[idx] = feats[rowbase * CK + idx];
    __syncthreads();

    // ---- subtile 0 (rows 0..15): A fragments in registers ----
    // ISA layout: lane(h, m): VGPR0 = A[m][k0+2h], VGPR1 = A[m][k0+2h+1]
    v2f a0_frags[NKS];
#pragma unroll
    for (int kk = 0; kk < NKS; ++kk)
        a0_frags[kk] = *(const v2f*)&smem.a[lm * CK + kk * 4 + 2 * half];

    // ---- subtile 1 (rows 16..31): A fragments in LDS, fragment order (bank-conflict-free) ----
    for (int e = tid; e < NKS * 32; e += 256) {
        const int kk = e >> 5, l = e & 31, hh = l >> 4, mm = l & 15;
        a1f[e] = *(const v2f*)&smem.a[(16 + mm) * CK + kk * 4 + 2 * hh];
    }

    int rl[2][8];
#pragma unroll
    for (int g = 0; g < 2; ++g)
#pragma unroll
        for (int r = 0; r < 8; ++r)
            rl[g][r] = labels[rowbase + g * 16 + r + 8 * half];  // slot (g,r) -> row g*16+r+8*half

    float mrow[2][8], srow[2][8], prow[2][8], pcnt[2][8];
#pragma unroll
    for (int g = 0; g < 2; ++g)
#pragma unroll
        for (int r = 0; r < 8; ++r) {
            mrow[g][r] = -INFINITY; srow[g][r] = 0.0f; prow[g][r] = 0.0f; pcnt[g][r] = 0.0f;
        }
    __syncthreads();   // all waves done reading raw A (and a1f built) before B staging reuses it

    // per-wave B buffer base as a raw LDS byte offset (async-load VDST operand)
    const uint32_t bwbase = (uint32_t)(uintptr_t)&smem.b[wv][0][0];
    const int sr = lane >> 3;                  // staging row sub-index 0..3
    const int sc = (lane & 7) * 4;             // staging col 0,4,...,28

    // async-stage one 16x32 B chunk (tile tt, k-chunk kcc) into buffer bufIdx (4 x b128 per lane)
    auto stage = [&](int tt, int kcc, int bufIdx) {
        asm volatile("s_wait_dscnt 0" ::: "memory");     // WAR: prior ds reads of this buffer done
        const float* src = fs + ((wv + tt * NW) * 16) * CK + kcc * KCH;
#pragma unroll
        for (int p = 0; p < 4; ++p) {
            const int r = p * 4 + sr;                    // 0..15
            const float*   g  = src + r * CK + sc;
            const uint32_t ld = bwbase + (uint32_t)((bufIdx * (16 * BPAD) + r * BPAD + sc) * 4);
            asm volatile("global_load_async_to_lds_b128 %0, %1, off"
                         :: "v"(ld), "v"(g) : "memory");
        }
    };

    // ---- software-pipelined sweep: prefetch chunk i+1 while WMMAs consume chunk i ----
    stage(0, 0, 0);                                      // prologue
    for (int t = 0; t < TPW; ++t) {
        const int colbase = (wv + t * NW) * 16;
        const int lbl     = labels_s[colbase + lm];      // this lane's column label

        v8f acc[2];
        acc[0] = (v8f){0.f, 0.f, 0.f, 0.f, 0.f, 0.f, 0.f, 0.f};
        acc[1] = (v8f){0.f, 0.f, 0.f, 0.f, 0.f, 0.f, 0.f, 0.f};

#pragma unroll
        for (int kc = 0; kc < NKC; ++kc) {
            // prefetch next chunk (wraps to chunk 0 at the end: valid addresses, result unused)
            const int tn = (kc == NKC - 1) ? ((t + 1 == TPW) ? 0 : t + 1) : t;
            stage(tn, (kc + 1) & (NKC - 1), (kc + 1) & 1);
            // async completions are in-order: cnt<=4 => the 4 oldest (current chunk) landed
            asm volatile("s_wait_asynccnt 4" ::: "memory");

            // 8 WMMA k-steps; each B fragment feeds both 16-row subtiles
            // B layout: lane(h, n): VGPR0 = B[k0+2h][n] = fs[colbase+n][k0+2h]
#pragma unroll
            for (int ks = 0; ks < 8; ++ks) {
                const v2f b  = *(const v2f*)&smem.b[wv][kc & 1][lm * BPAD + ks * 4 + 2 * half];
                const v2f a1 = a1f[(kc * 8 + ks) * 32 + lane];
                acc[0] = __builtin_amdgcn_wmma_f32_16x16x4_f32(
                            false, a0_frags[kc * 8 + ks], false, b, (short)0, acc[0], false, false);
                acc[1] = __builtin_amdgcn_wmma_f32_16x16x4_f32(
                            false, a1, false, b, (short)0, acc[1], false, false);
            }
        }

        // ---- online softmax + positive accumulation (slot (g,r) -> row g*16+r+8*half, col lm) ----
#pragma unroll
        for (int g = 0; g < 2; ++g)
#pragma unroll
            for (int r = 0; r < 8; ++r) {
                const float v = acc[g][r] * INV_TEMP;
                online_update(mrow[g][r], srow[g][r], v);
                const bool pos = (lbl == rl[g][r]);
                prow[g][r] += pos ? v : 0.0f;
                pcnt[g][r] += pos ? 1.0f : 0.0f;
            }
    }

    // ---- butterfly-reduce each row across its 16-lane half ----
#pragma unroll
    for (int g = 0; g < 2; ++g)
#pragma unroll
        for (int r = 0; r < 8; ++r) {
            float m = mrow[g][r], s = srow[g][r], p = prow[g][r], c = pcnt[g][r];
#pragma unroll
            for (int off = 1; off <= 8; off <<= 1) {
                const float mo = __shfl_xor(m, off, 32);
                const float so = __shfl_xor(s, off, 32);
                p += __shfl_xor(p, off, 32);
                c += __shfl_xor(c, off, 32);
                pair_merge(m, s, mo, so);
            }
            if (lm == 0)
                lds_red[wv][g * 16 + r + 8 * half] = make_float4(m, s, p, c);
        }
    __syncthreads();

    // ---- wave 0: lane = row (32 rows), merge the 8 waves, per-row loss, reduce to one value ----
    if (wv == 0) {
        float m = -INFINITY, s = 0.0f, p = 0.0f, c = 0.0f;
#pragma unroll
        for (int w = 0; w < NW; ++w) {
            const float4 v = lds_red[w][lane];
            pair_merge(m, s, v.x, v.y);
            p += v.z; c += v.w;
        }
        float loss = (m + __logf(s)) - p / c;   // lse - pos_sum / P
#pragma unroll
        for (int off = 1; off <= 16; off <<= 1)
            loss += __shfl_xor(loss, off, 32);
        if (lane == 0)
            block_loss[blockIdx.x] = loss;
    }
}

// Deterministic final reduction: 128 block losses -> scalar mean over B
__global__ __launch_bounds__(128) void lp_loss_reduce(
    const float* __restrict__ block_loss, float* __restrict__ out)
{
    __shared__ float sm[128];
    sm[threadIdx.x] = block_loss[threadIdx.x];
    __syncthreads();
#pragma unroll
    for (int s = 64; s > 0; s >>= 1) {
        if (threadIdx.x < s) sm[threadIdx.x] += sm[threadIdx.x + s];
        __syncthreads();
    }
    if (threadIdx.x == 0) out[0] = sm[0] / (float)B_ROWS;
}

extern "C" void kernel_launch(void* const* d_in, const int* in_sizes, int n_in,
                              void* d_out, int out_size, void* d_ws, size_t ws_size,
                              hipStream_t stream) {
    const float* feats    = (const float*)d_in[0];   // [4096,128]
    const float* feats_s  = (const float*)d_in[1];   // [4096,4,128] -> flat [16384,128]
    const int*   labels   = (const int*)d_in[2];     // [4096]
    const int*   labels_s = (const int*)d_in[3];     // [16384]
    float*       ws       = (float*)d_ws;            // 128 floats of scratch
    float*       out      = (float*)d_out;           // scalar

    lp_loss_main<<<NBLOCKS, 256, 0, stream>>>(feats, feats_s, labels, labels_s, ws);
    lp_loss_reduce<<<1, 128, 0, stream>>>(ws, out);
}